// MPNN_37615323578487
// MI455X (gfx1250) — compile-verified
//
#include <hip/hip_runtime.h>

// MPNN layer for MI455X (gfx1250, wave32).
// K1: feat_src = lrelu(feat @ W1^T + b1)     [WMMA f32 16x16x4] -> d_out (temp)
//     + zero ft accumulator in d_ws
// K2: ft[dst] += feat_src[src] * belta/dist  [wave/edge, global_atomic_add_f32]
// K3: out = lrelu((ft + (feat.att_w)*feat) @ W2^T + b2)  [WMMA] -> d_out

#define N_NODES 50000
#define DFEAT   64
#define SLOPE   0.2f
#define LDS_PITCH 68   // 64 + 4 pad: stride mod 64 banks == 4 -> conflict-free b64 reads

typedef __attribute__((ext_vector_type(2))) float v2f;
typedef __attribute__((ext_vector_type(4))) float v4f;
typedef __attribute__((ext_vector_type(8))) float v8f;

__device__ __forceinline__ float lrelu(float x) { return x > 0.0f ? x : SLOPE * x; }

// ---------------------------------------------------------------------------
// Kernel 1: fc1 + leaky_relu, plus zeroing of the ft accumulator.
// One wave (32 lanes) computes a 16x64 output tile via 64 x V_WMMA_F32_16X16X4_F32.
// ---------------------------------------------------------------------------
__global__ __launch_bounds__(256) void mpnn_fc1_kernel(
    const float* __restrict__ feat, const float* __restrict__ W1,
    const float* __restrict__ b1, float* __restrict__ feat_src,
    float* __restrict__ ft)
{
    __shared__ float ldsW[DFEAT * LDS_PITCH];
    const int tid = threadIdx.x;

    // Stage W1 (64x64 f32, 16 KB) into LDS with padded pitch.
    for (int i = tid; i < (DFEAT * DFEAT / 4); i += 256) {
        v4f v = ((const v4f*)W1)[i];
        const int r = (i * 4) / DFEAT;
        const int c = (i * 4) % DFEAT;
        *(v4f*)(&ldsW[r * LDS_PITCH + c]) = v;
    }
    __syncthreads();

    const int lane = tid & 31;
    const int wave = tid >> 5;
    const int m    = lane & 15;   // A row within tile / D col within tile
    const int h    = lane >> 4;   // half of wave
    const int row0 = (blockIdx.x * 8 + wave) * 16;

    // Zero ft rows owned by this tile (16 rows x 64 f32 = 256 float4).
    {
        const v4f z = {0.f, 0.f, 0.f, 0.f};
        const int q0 = row0 * (DFEAT / 4);
        #pragma unroll
        for (int i = 0; i < 8; ++i) {
            const int q = q0 + lane + 32 * i;
            if (q < N_NODES * (DFEAT / 4)) ((v4f*)ft)[q] = z;
        }
    }

    // Load A tile (16x64) in WMMA A layout: lane holds feat[rowA][4s+2h .. +1].
    const int rowA = (row0 + m < N_NODES) ? (row0 + m) : (N_NODES - 1);
    const float* fr = feat + (size_t)rowA * DFEAT;
    v2f a[16];
    #pragma unroll
    for (int s = 0; s < 16; ++s) a[s] = *(const v2f*)(fr + 4 * s + 2 * h);

    const v8f zero8 = {0.f, 0.f, 0.f, 0.f, 0.f, 0.f, 0.f, 0.f};
    v8f acc[4];
    #pragma unroll
    for (int t = 0; t < 4; ++t) acc[t] = zero8;

    // D = A x B, B[k][n] = W1[n][k]; lane's B pair {W1[n][k],W1[n][k+1]} is a
    // contiguous 8B chunk of the staged weight row.
    #pragma unroll
    for (int s = 0; s < 16; ++s) {
        #pragma unroll
        for (int t = 0; t < 4; ++t) {
            const int n = 16 * t + m;
            const v2f b = *(const v2f*)(&ldsW[n * LDS_PITCH + 4 * s + 2 * h]);
            acc[t] = __builtin_amdgcn_wmma_f32_16x16x4_f32(
                false, a[s], false, b, (short)0, acc[t], false, false);
        }
    }

    // Epilogue: bias + leaky relu, scatter to feat_src (row = row0 + v + 8h).
    #pragma unroll
    for (int t = 0; t < 4; ++t) {
        const int n = 16 * t + m;
        const float bias = b1[n];
        #pragma unroll
        for (int v = 0; v < 8; ++v) {
            const int row = row0 + v + 8 * h;
            if (row < N_NODES)
                feat_src[(size_t)row * DFEAT + n] = lrelu(acc[t][v] + bias);
        }
    }
}

// ---------------------------------------------------------------------------
// Kernel 2: edge gather/scale/scatter-add. One wave per 32-edge chunk.
// Lane i loads metadata for edge (chunk*32+i) coalesced; the wave then walks
// the 32 edges, broadcasting (w,src,dst) via __shfl and doing a contiguous
// 256B gather (float2/lane) + 2 f32 atomics per lane.
// ---------------------------------------------------------------------------
__global__ __launch_bounds__(256) void mpnn_edge_kernel(
    const float* __restrict__ feat_src, const float* __restrict__ dist,
    const int* __restrict__ src, const int* __restrict__ dst,
    const float* __restrict__ belta, float* __restrict__ ft, int E)
{
    const int lane = threadIdx.x & 31;
    const int wid  = (blockIdx.x * blockDim.x + threadIdx.x) >> 5;
    const int nw   = (gridDim.x * blockDim.x) >> 5;
    const float bv = belta[0];
    const int nchunks = (E + 31) >> 5;

    for (int c = wid; c < nchunks; c += nw) {
        const int eid   = c * 32 + lane;
        const bool ok   = eid < E;
        const float w_l = ok ? bv / dist[eid] : 0.0f;  // w=0 => no-op adds on tail
        const int   s_l = ok ? src[eid] : 0;
        const int   d_l = ok ? dst[eid] : 0;
        #pragma unroll
        for (int j = 0; j < 32; ++j) {
            const float w = __shfl(w_l, j);
            const int   s = __shfl(s_l, j);
            const int   d = __shfl(d_l, j);
            const v2f v = *(const v2f*)(feat_src + (size_t)s * DFEAT + 2 * lane);
            float* o = ft + (size_t)d * DFEAT + 2 * lane;
            atomicAdd(o,     v.x * w);
            atomicAdd(o + 1, v.y * w);
        }
    }
}

// ---------------------------------------------------------------------------
// Kernel 3: rst = lrelu((ft + e*feat) @ W2^T + b2), e = feat . att_w per row.
// Same WMMA tiling as K1; e computed from the A-tile registers with a
// cross-half __shfl_xor(16) reduction (wave32).
// ---------------------------------------------------------------------------
__global__ __launch_bounds__(256) void mpnn_fc2_kernel(
    const float* __restrict__ feat, const float* __restrict__ ft,
    const float* __restrict__ W2, const float* __restrict__ b2,
    const float* __restrict__ att_w, float* __restrict__ out)
{
    __shared__ float ldsW[DFEAT * LDS_PITCH];
    __shared__ float ldsA[DFEAT];
    const int tid = threadIdx.x;

    for (int i = tid; i < (DFEAT * DFEAT / 4); i += 256) {
        v4f v = ((const v4f*)W2)[i];
        const int r = (i * 4) / DFEAT;
        const int c = (i * 4) % DFEAT;
        *(v4f*)(&ldsW[r * LDS_PITCH + c]) = v;
    }
    if (tid < DFEAT) ldsA[tid] = att_w[tid];
    __syncthreads();

    const int lane = tid & 31;
    const int wave = tid >> 5;
    const int m    = lane & 15;
    const int h    = lane >> 4;
    const int row0 = (blockIdx.x * 8 + wave) * 16;
    const int rowA = (row0 + m < N_NODES) ? (row0 + m) : (N_NODES - 1);
    const float* fr = feat + (size_t)rowA * DFEAT;
    const float* tr = ft   + (size_t)rowA * DFEAT;

    v2f fa[16], aa[16];
    float ep = 0.0f;
    #pragma unroll
    for (int s = 0; s < 16; ++s) {
        const int k = 4 * s + 2 * h;
        fa[s] = *(const v2f*)(fr + k);
        aa[s] = *(const v2f*)(tr + k);
        const v2f aw = *(const v2f*)(&ldsA[k]);
        ep += fa[s].x * aw.x + fa[s].y * aw.y;
    }
    // lanes L and L^16 hold complementary k-subsets of the same row m
    const float e = ep + __shfl_xor(ep, 16);

    #pragma unroll
    for (int s = 0; s < 16; ++s) {
        aa[s].x += e * fa[s].x;
        aa[s].y += e * fa[s].y;
    }

    const v8f zero8 = {0.f, 0.f, 0.f, 0.f, 0.f, 0.f, 0.f, 0.f};
    v8f acc[4];
    #pragma unroll
    for (int t = 0; t < 4; ++t) acc[t] = zero8;

    #pragma unroll
    for (int s = 0; s < 16; ++s) {
        #pragma unroll
        for (int t = 0; t < 4; ++t) {
            const int n = 16 * t + m;
            const v2f b = *(const v2f*)(&ldsW[n * LDS_PITCH + 4 * s + 2 * h]);
            acc[t] = __builtin_amdgcn_wmma_f32_16x16x4_f32(
                false, aa[s], false, b, (short)0, acc[t], false, false);
        }
    }

    #pragma unroll
    for (int t = 0; t < 4; ++t) {
        const int n = 16 * t + m;
        const float bias = b2[n];
        #pragma unroll
        for (int v = 0; v < 8; ++v) {
            const int row = row0 + v + 8 * h;
            if (row < N_NODES)
                out[(size_t)row * DFEAT + n] = lrelu(acc[t][v] + bias);
        }
    }
}

// ---------------------------------------------------------------------------
extern "C" void kernel_launch(void* const* d_in, const int* in_sizes, int n_in,
                              void* d_out, int out_size, void* d_ws, size_t ws_size,
                              hipStream_t stream) {
    const float* feat  = (const float*)d_in[0];
    const float* dist  = (const float*)d_in[1];
    const int*   src   = (const int*)d_in[2];
    const int*   dst   = (const int*)d_in[3];
    const float* W1    = (const float*)d_in[4];
    const float* b1    = (const float*)d_in[5];
    const float* W2    = (const float*)d_in[6];
    const float* b2    = (const float*)d_in[7];
    const float* att_w = (const float*)d_in[8];
    const float* belta = (const float*)d_in[9];
    const int E = in_sizes[1];

    float* out = (float*)d_out;          // feat_src lives here between K1 and K2
    float* ft  = (float*)d_ws;           // N*64 f32 accumulator (12.8 MB)

    const int tiles  = (N_NODES + 15) / 16;   // 3125 wave-tiles
    const int blocks = (tiles + 7) / 8;       // 8 waves (tiles) per 256-thread block

    mpnn_fc1_kernel<<<blocks, 256, 0, stream>>>(feat, W1, b1, out, ft);
    mpnn_edge_kernel<<<1024, 256, 0, stream>>>(out, dist, src, dst, belta, ft, E);
    mpnn_fc2_kernel<<<blocks, 256, 0, stream>>>(feat, ft, W2, b2, att_w, out);
}